// GaussianLSS_34986803593228
// MI455X (gfx1250) — compile-verified
//
#include <hip/hip_runtime.h>

#define IMG    200
#define TILE   16
#define CAP    1024
#define THRESHV 0.05f

typedef float v2f __attribute__((ext_vector_type(2)));
typedef float v8f __attribute__((ext_vector_type(8)));

__device__ __forceinline__ float evalAlpha(float u, float v, float cu, float cv,
                                           float ca, float cb, float cc, float op,
                                           float pu, float pv) {
    float du = pu - u, dv = pv - v;
    float power = -0.5f * (ca * du * du + cc * dv * dv) - cb * du * dv;
    float al = fminf(0.99f, op * expf(power));
    bool valid = (fabsf(pu - cu) <= 3.0f) && (fabsf(pv - cv) <= 3.0f) &&
                 (al >= (1.0f / 255.0f));
    return valid ? al : 0.0f;
}

__device__ __forceinline__ void storeRow(float* __restrict__ out, int b, int ch,
                                         int pv, int puBase, const v8f& c) {
    if (pv >= IMG) return;
    size_t off = (((size_t)(b * 32 + ch)) * IMG + pv) * IMG + puBase;
    float4 lo = make_float4(c[0], c[1], c[2], c[3]);
    float4 hi = make_float4(c[4], c[5], c[6], c[7]);
    *(float4*)(out + off)     = lo;
    *(float4*)(out + off + 4) = hi;
}

// One workgroup = one 16x16 pixel tile of one batch image.
// Phase 1: 256 threads scan gaussians, push overlapping ones (precomputed conic)
//          into an LDS list.  Phase 2: 8 waves consume the list; wave w owns
//          pixel rows 2w, 2w+1; per 4 gaussians it builds alpha A-fragments
//          (16 px x 4 g) and feature B-fragments (4 g x 16 ch) and issues
//          4x v_wmma_f32_16x16x4_f32 into f32 accumulators (16 px x 32 ch).
__global__ __launch_bounds__(256)
void raster_tile_kernel(const float* __restrict__ feats,
                        const float* __restrict__ means,
                        const float* __restrict__ cov,
                        const float* __restrict__ opac,
                        float* __restrict__ out, int G) {
    __shared__ float su[CAP], sv[CAP], scu[CAP], scv[CAP];
    __shared__ float sca[CAP], scb[CAP], scc[CAP], sop[CAP];
    __shared__ int   sg[CAP];
    __shared__ int   scount;

    const int b   = blockIdx.z;
    const int tu0 = blockIdx.x * TILE;   // fast image dim (pu / width)
    const int tv0 = blockIdx.y * TILE;   // slow image dim (pv / height)
    const int tid  = threadIdx.x;
    const int lane = tid & 31;
    const int wv   = tid >> 5;           // wave id 0..7

    const int   pxl   = lane & 15;       // M index within 16-pixel group
    const int   hi    = lane >> 4;       // lane half -> K group {0,1}->K{0,1}, {2,3}
    const float pu_f  = (float)(tu0 + pxl);
    const float pv0_f = (float)(tv0 + 2 * wv);
    const float pv1_f = pv0_f + 1.0f;

    v8f c00 = {}; v8f c01 = {}; v8f c10 = {}; v8f c11 = {};

    if (tid == 0) scount = 0;
    __syncthreads();

    const size_t gbase = (size_t)b * G;

    int base = 0;
    while (base < G) {
        // ---- scan pass: one gaussian per thread ----
        int g = base + tid;
        if (g < G) {
            float op = opac[gbase + g];
            if (op > THRESHV) {
                float m0 = means[(gbase + g) * 3 + 0];
                float m1 = means[(gbase + g) * 3 + 1];
                float u = 100.0f - 2.0f * m1;     // -SH*y + IMG/2
                float v = 100.0f - 2.0f * m0;     // -SW*x + IMG/2
                float cu = rintf(u), cv = rintf(v);  // round-half-even (jnp.round)
                if (cu + 3.0f >= (float)tu0 && cu - 3.0f <= (float)(tu0 + TILE - 1) &&
                    cv + 3.0f >= (float)tv0 && cv - 3.0f <= (float)(tv0 + TILE - 1)) {
                    float c0 = cov[(gbase + g) * 6 + 0];
                    float c1 = cov[(gbase + g) * 6 + 1];
                    float c3 = cov[(gbase + g) * 6 + 3];
                    float A  = 4.0f * c3 + 0.3f;
                    float Bc = 4.0f * c1;
                    float C  = 4.0f * c0 + 0.3f;
                    float inv = 1.0f / (A * C - Bc * Bc);
                    int pos = atomicAdd(&scount, 1);   // never exceeds CAP-1 by flush policy
                    su[pos] = u;  sv[pos] = v;  scu[pos] = cu;  scv[pos] = cv;
                    sca[pos] = C * inv;  scb[pos] = -Bc * inv;  scc[pos] = A * inv;
                    sop[pos] = op;  sg[pos] = g;
                }
            }
        }
        base += 256;
        __syncthreads();

        // ---- flush (consume list) when nearly full or done ----
        if (scount > CAP - 256 || base >= G) {
            int n = scount; if (n > CAP) n = CAP;
            for (int kk = 0; kk < n; kk += 4) {
                int e0 = kk + 2 * hi;
                int e1 = e0 + 1;
                int e0c = (e0 < n) ? e0 : (n - 1);
                int e1c = (e1 < n) ? e1 : (n - 1);
                float m0v = (e0 < n) ? 1.0f : 0.0f;
                float m1v = (e1 < n) ? 1.0f : 0.0f;

                float u0 = su[e0c], v0 = sv[e0c], cu0 = scu[e0c], cv0 = scv[e0c];
                float a0 = sca[e0c], b0 = scb[e0c], q0 = scc[e0c], o0 = sop[e0c];
                float u1 = su[e1c], v1 = sv[e1c], cu1 = scu[e1c], cv1 = scv[e1c];
                float a1 = sca[e1c], b1 = scb[e1c], q1 = scc[e1c], o1 = sop[e1c];

                // A fragments: 16 pixels x 4 gaussians (rows pv0, pv1), tail masked
                v2f ar0, ar1;
                ar0[0] = m0v * evalAlpha(u0, v0, cu0, cv0, a0, b0, q0, o0, pu_f, pv0_f);
                ar0[1] = m1v * evalAlpha(u1, v1, cu1, cv1, a1, b1, q1, o1, pu_f, pv0_f);
                ar1[0] = m0v * evalAlpha(u0, v0, cu0, cv0, a0, b0, q0, o0, pu_f, pv1_f);
                ar1[1] = m1v * evalAlpha(u1, v1, cu1, cv1, a1, b1, q1, o1, pu_f, pv1_f);

                // B fragments: 4 gaussians x 16 channels (ch 0-15 and 16-31)
                size_t f0 = (gbase + (size_t)sg[e0c]) * 32;
                size_t f1 = (gbase + (size_t)sg[e1c]) * 32;
                v2f blo, bhi;
                blo[0] = feats[f0 + pxl];       blo[1] = feats[f1 + pxl];
                bhi[0] = feats[f0 + 16 + pxl];  bhi[1] = feats[f1 + 16 + pxl];

                c00 = __builtin_amdgcn_wmma_f32_16x16x4_f32(false, ar0, false, blo,
                                                            (short)0, c00, false, false);
                c01 = __builtin_amdgcn_wmma_f32_16x16x4_f32(false, ar0, false, bhi,
                                                            (short)0, c01, false, false);
                c10 = __builtin_amdgcn_wmma_f32_16x16x4_f32(false, ar1, false, blo,
                                                            (short)0, c10, false, false);
                c11 = __builtin_amdgcn_wmma_f32_16x16x4_f32(false, ar1, false, bhi,
                                                            (short)0, c11, false, false);
            }
            __syncthreads();
            if (tid == 0) scount = 0;
            __syncthreads();
        }
    }

    // ---- store: lane holds channel (pxl [+16]) for 8 consecutive pu ----
    int puBase = tu0 + 8 * hi;
    int pv0 = tv0 + 2 * wv, pv1 = pv0 + 1;
    if (puBase + 7 < IMG) {
        storeRow(out, b, pxl,      pv0, puBase, c00);
        storeRow(out, b, pxl + 16, pv0, puBase, c01);
        storeRow(out, b, pxl,      pv1, puBase, c10);
        storeRow(out, b, pxl + 16, pv1, puBase, c11);
    }
}

__global__ __launch_bounds__(1024)
void count_kernel(const float* __restrict__ opac, float* __restrict__ cnt,
                  int total, float invB) {
    __shared__ float red[1024];
    int t = threadIdx.x;
    float s = 0.0f;
    for (int i = t; i < total; i += 1024)
        s += (opac[i] > THRESHV) ? 1.0f : 0.0f;
    red[t] = s;
    __syncthreads();
    for (int off = 512; off > 0; off >>= 1) {
        if (t < off) red[t] += red[t + off];
        __syncthreads();
    }
    if (t == 0) cnt[0] = red[0] * invB;
}

extern "C" void kernel_launch(void* const* d_in, const int* in_sizes, int n_in,
                              void* d_out, int out_size, void* d_ws, size_t ws_size,
                              hipStream_t stream) {
    const float* feats = (const float*)d_in[0];  // (B,G,32)
    const float* means = (const float*)d_in[1];  // (B,G,3)
    const float* cov   = (const float*)d_in[2];  // (B,G,6)
    const float* opac  = (const float*)d_in[3];  // (B,G,1)
    float* out = (float*)d_out;                  // (B,32,200,200) flat + 1 count

    int B = (out_size - 1) / (32 * IMG * IMG);   // = 2
    if (B < 1) B = 1;
    int BG = in_sizes[3];                        // B*G
    int G  = BG / B;

    dim3 grid((IMG + TILE - 1) / TILE, (IMG + TILE - 1) / TILE, B);
    raster_tile_kernel<<<grid, 256, 0, stream>>>(feats, means, cov, opac, out, G);

    float* cnt = out + (size_t)out_size - 1;
    count_kernel<<<1, 1024, 0, stream>>>(opac, cnt, BG, 1.0f / (float)B);
}